// ARDG_2946347565852
// MI455X (gfx1250) — compile-verified
//
#include <hip/hip_runtime.h>
#include <stdint.h>

// ---------------- problem constants ----------------
#define BB   256
#define NN   128
#define DXC  8
#define DEC  5
#define DX1  9
#define DE1  6
#define NPOS (NN*NN)            // 16384 positions per batch
#define EROW (NN*DE1)           // 768 floats per j-row of E
#define EBATCH ((size_t)NPOS*DE1) // 98304 floats per batch
#define SLICES 8
#define ROWS_PER_SLICE (NN/SLICES) // 16 rows -> 48KB tile
#define MAXK 128
#define NX_OUT ((size_t)BB*NN*DX1) // 294912 floats, X_out precedes E_out in d_out

typedef __attribute__((ext_vector_type(2))) float        v2f;
typedef __attribute__((ext_vector_type(8))) float        v8f;
typedef __attribute__((ext_vector_type(4))) unsigned int v4u;
typedef __attribute__((ext_vector_type(8))) int          v8i;
typedef __attribute__((ext_vector_type(4))) int          v4i;

// ---------------------------------------------------------------------------
// K1: stream E[b] slice (16 rows, 48KB) into LDS via the Tensor Data Mover,
// compute per-position argmax class (u8 -> ws), and reduce three partials
// (bE = sum pair*E.we, Swe = triu sum pair*we[cls], cntMaskedE) with
// V_WMMA_F32_16X16X4_F32 (B = ones -> exact f32 row-sum reduction).
// Per-slice results go to fixed ws slots (deterministic, no atomics).
// ---------------------------------------------------------------------------
__global__ __launch_bounds__(256)
void ardg_k1_epass(const float* __restrict__ E,
                   const unsigned char* __restrict__ nmask,
                   const float* __restrict__ we,
                   unsigned char* __restrict__ ec,
                   float* __restrict__ acc /* [BB*SLICES][4] */) {
  const int b = blockIdx.x, s = blockIdx.y, tid = threadIdx.x;
  const int j0 = s * ROWS_PER_SLICE;

  __shared__ float sE[ROWS_PER_SLICE * EROW];   // 48KB tile of E
  __shared__ float red[3 * 256];
  __shared__ float red2[3 * 32];

  // ---- TDM: DMA the 16-row slice (12288 floats = 6144 x 8B) into LDS ----
  if ((tid >> 5) == 0) {   // wave-uniform guard: only wave 0 issues the TDM op
    unsigned long long ga =
        (unsigned long long)(uintptr_t)(E + (size_t)b * EBATCH + (size_t)j0 * EROW);
    unsigned int lds = (unsigned int)(uintptr_t)&sE[0]; // low 32 bits = LDS offset

    v4u g0;
    g0.x = 1u;                                   // count=1, user mode, no gather
    g0.y = lds;                                  // lds_addr
    g0.z = (unsigned int)(ga & 0xFFFFFFFFull);   // global_addr[31:0]
    g0.w = (unsigned int)((ga >> 32) & 0x01FFFFFFull) | (2u << 30); // addr[56:32]|type=2

    const unsigned int dim0 = 6144u;  // tensor_dim0 in 8B units
    const unsigned int dim1 = 1u;
    const unsigned int tile0 = 6144u; // tile_dim0 (1-D copy)
    v8i g1;
    g1[0] = (int)(3u << 16);                                  // data_size=8B
    g1[1] = (int)((dim0 & 0xFFFFu) << 16);                    // dim0[15:0] @ bits63:48
    g1[2] = (int)(((dim0 >> 16) & 0xFFFFu) | ((dim1 & 0xFFFFu) << 16));
    g1[3] = (int)(((dim1 >> 16) & 0xFFFFu) | (tile0 << 16));  // tile_dim0 @ 127:112
    g1[4] = 0;                                                // tile_dim1=tile_dim2=0
    g1[5] = (int)dim0;                                        // tensor_dim0_stride lo
    g1[6] = 0;
    g1[7] = 0;
    v4i gz4 = {0, 0, 0, 0};
    v8i gz8 = {0, 0, 0, 0, 0, 0, 0, 0};
    // 6-arg form on this toolchain: (g0, g1, g2, g3, g4, cpol)
    __builtin_amdgcn_tensor_load_to_lds(g0, g1, gz4, gz4, gz8, 0);
    __builtin_amdgcn_s_wait_tensorcnt(0);
  }
  __syncthreads();

  float w6[DE1];
#pragma unroll
  for (int c = 0; c < DE1; ++c) w6[c] = we[c];
  const unsigned char* nmb = nmask + (size_t)b * NN;

  float pbE = 0.f, pSwe = 0.f, pCnt = 0.f;
#pragma unroll
  for (int p = 0; p < (ROWS_PER_SLICE * NN) / 256; ++p) {   // 8 positions/thread
    int pos = tid + p * 256;
    int jj = pos >> 7, k = pos & (NN - 1);
    int j = j0 + jj;
    const float* ev = &sE[jj * EROW + k * DE1];
    int cls = 0; float bv = ev[0]; float dot = ev[0] * w6[0];
#pragma unroll
    for (int c = 1; c < DE1; ++c) {
      float v = ev[c]; dot += v * w6[c];
      if (v > bv) { bv = v; cls = c; }           // first-max tie-break (jnp.argmax)
    }
    float pr = (nmb[j] && nmb[k]) ? 1.f : 0.f;
    pbE += pr * dot;
    if (j < k) {
      pSwe += pr * w6[cls];
      pCnt += (cls == DEC) ? pr : 0.f;           // masked edge (class 5) in triu
    }
    ec[(size_t)b * NPOS + (size_t)j * NN + k] = (unsigned char)cls;
  }
  red[0 * 256 + tid] = pbE;
  red[1 * 256 + tid] = pSwe;
  red[2 * 256 + tid] = pCnt;
  __syncthreads();

  // ---- WMMA reduction: C += A(16x4 f32 chunk) x B(ones) -> row sums ----
  if ((tid >> 5) == 0) {
    int lane = tid;
    v2f ones; ones.x = 1.f; ones.y = 1.f;
    int off = (lane < 16) ? lane * 4 : (lane - 16) * 4 + 2; // A 16x4 f32 VGPR layout
#pragma unroll
    for (int t = 0; t < 3; ++t) {
      v8f cacc = {};
#pragma unroll
      for (int ch = 0; ch < 4; ++ch) {           // 4 chunks x 64 = 256 partials
        const float* src = &red[t * 256 + ch * 64 + off];
        v2f a; a.x = src[0]; a.y = src[1];
        cacc = __builtin_amdgcn_wmma_f32_16x16x4_f32(false, a, false, ones,
                                                     (short)0, cacc, false, false);
      }
      float ssum = cacc[0] + cacc[1] + cacc[2] + cacc[3] +
                   cacc[4] + cacc[5] + cacc[6] + cacc[7];
      red2[t * 32 + lane] = ssum;                // lane0 + lane16 carry column 0
    }
  }
  __syncthreads();
  if (tid == 0) {
    int slot = (b * SLICES + s) * 4;
#pragma unroll
    for (int t = 0; t < 3; ++t)
      acc[slot + t] = red2[t * 32 + 0] + red2[t * 32 + 16];
  }
}

// ---------------------------------------------------------------------------
// K2: one block (128 threads) per batch. X-side reductions, temp, d-tables,
// top-k Gumbel selection (value desc, index asc tie-break == double argsort),
// guided Gumbel-argmax predictions, class overwrites in ws.
// ---------------------------------------------------------------------------
__global__ __launch_bounds__(128)
void ardg_k2_finalize(const float* __restrict__ X,
                      const unsigned char* __restrict__ nmask,
                      const float* __restrict__ y,
                      const float* __restrict__ logits_x,
                      const float* __restrict__ logits_e,
                      const float* __restrict__ g_edge,
                      const float* __restrict__ g_node,
                      const float* __restrict__ g_cat_e,
                      const float* __restrict__ g_cat_x,
                      const float* __restrict__ wx,
                      const float* __restrict__ we,
                      const float* __restrict__ wy,
                      const int* __restrict__ step_ptr,
                      const float* __restrict__ acc,
                      unsigned char* __restrict__ ec,
                      unsigned char* __restrict__ xc) {
  const int b = blockIdx.x, tid = threadIdx.x;
  const unsigned char* nmb = nmask + (size_t)b * NN;

  __shared__ float s_red[NN];
  __shared__ int   s_xcls[NN];
  __shared__ float s_val[NN];
  __shared__ int   s_idx[NN];
  __shared__ int   s_pick[MAXK];
  __shared__ int   s_pickn[MAXK];
  __shared__ float s_de[DEC];
  __shared__ float s_dx[DXC];
  __shared__ int   s_ke, s_kn;

  // ---- per-node: argmax class, Sx, maskedN, numNodes ----
  const int j = tid;
  const float* xr = X + ((size_t)b * NN + j) * DX1;
  int xcls = 0; float bv = xr[0];
#pragma unroll
  for (int c = 1; c < DX1; ++c) { float v = xr[c]; if (v > bv) { bv = v; xcls = c; } }
  s_xcls[j] = xcls;
  xc[(size_t)b * NN + j] = (unsigned char)xcls;   // base class; picks overwritten below
  float nmj = nmb[j] ? 1.f : 0.f;

  float SX = 0.f, MN = 0.f, NNODES = 0.f;
  s_red[tid] = nmj * wx[xcls]; __syncthreads();
  if (tid == 0) { float t = 0.f; for (int i = 0; i < NN; ++i) t += s_red[i]; s_red[0] = t; }
  __syncthreads(); SX = s_red[0]; __syncthreads();
  s_red[tid] = (nmj > 0.f && xcls == DXC) ? 1.f : 0.f; __syncthreads();
  if (tid == 0) { float t = 0.f; for (int i = 0; i < NN; ++i) t += s_red[i]; s_red[0] = t; }
  __syncthreads(); MN = s_red[0]; __syncthreads();
  s_red[tid] = nmj; __syncthreads();
  if (tid == 0) { float t = 0.f; for (int i = 0; i < NN; ++i) t += s_red[i]; s_red[0] = t; }
  __syncthreads(); NNODES = s_red[0]; __syncthreads();

  if (tid == 0) {
    float bE = 0.f, Swe = 0.f, mE = 0.f;
    for (int s = 0; s < SLICES; ++s) {           // deterministic fixed-order sum
      const float* a = acc + (size_t)(b * SLICES + s) * 4;
      bE += a[0]; Swe += a[1]; mE += a[2];
    }
    float by = 0.f;
    for (int c = 0; c < 4; ++c) by += y[(size_t)b * 4 + c] * wy[c];
    int step = *step_ptr;
    int mEi = (int)(mE + 0.5f), mNi = (int)(MN + 0.5f);
    int ke = mEi < step ? mEi : step; if (ke > MAXK) ke = MAXK;
    int kn = mNi < step ? mNi : step; if (kn > MAXK) kn = MAXK;
    float num_elem = NNODES + NNODES * (NNODES - 1.f) * 0.5f;
    float temp = 0.5f * (1.f - (MN + mE) / num_elem);
    float we5 = we[DEC], wx8 = wx[DXC];
    for (int i = 0; i < DEC; ++i)
      s_de[i] = temp * (SX * (1.f / NN) + by +
                        (2.f / (float)NPOS) * (Swe - ke * we5 + ke * we[i]));
    for (int i = 0; i < DXC; ++i)
      s_dx[i] = temp * (bE * (1.f / (float)NPOS) + by +
                        (1.f / NN) * (SX - kn * wx8 + kn * wx[i]));
    s_ke = ke; s_kn = kn;
  }
  __syncthreads();
  const int ke = s_ke, kn = s_kn;

  // ---- edge top-k: k largest g_edge among absorbed triu positions ----
  for (int it = 0; it < ke; ++it) {
    float best = -3.0e38f; int bidx = 0x7FFFFFFF;
    for (int i = tid; i < NPOS; i += NN) {
      int jj = i >> 7, kk = i & (NN - 1);
      if (jj >= kk) continue;
      if (ec[(size_t)b * NPOS + i] != DEC) continue;
      if (!(nmb[jj] && nmb[kk])) continue;
      bool taken = false;
      for (int q = 0; q < it; ++q) if (s_pick[q] == i) { taken = true; break; }
      if (taken) continue;
      float v = g_edge[(size_t)b * NPOS + i];
      if (v > best || (v == best && i < bidx)) { best = v; bidx = i; }
    }
    s_val[tid] = best; s_idx[tid] = bidx; __syncthreads();
    if (tid == 0) {
      float bb = s_val[0]; int bi = s_idx[0];
      for (int t = 1; t < NN; ++t)
        if (s_val[t] > bb || (s_val[t] == bb && s_idx[t] < bi)) { bb = s_val[t]; bi = s_idx[t]; }
      s_pick[it] = bi;
    }
    __syncthreads();
  }

  // ---- edge predictions: argmax_c log(softmax(l)*e^{d_c}+1e-30)+g ----
  for (int p = tid; p < ke; p += NN) {
    int i = s_pick[p];
    int jj = i >> 7, kk = i & (NN - 1);
    const float* le = logits_e + ((size_t)b * NPOS + i) * DEC;
    const float* ge = g_cat_e + ((size_t)b * NPOS + i) * DEC;
    float m = le[0];
    for (int c = 1; c < DEC; ++c) m = fmaxf(m, le[c]);
    float den = 0.f;
    for (int c = 0; c < DEC; ++c) den += __expf(le[c] - m);
    int cls = 0; float bsc = -3.0e38f;
    for (int c = 0; c < DEC; ++c) {
      float pc = __expf(le[c] - m) / den;
      float sc = __logf(pc * __expf(s_de[c]) + 1e-30f) + ge[c];
      if (sc > bsc) { bsc = sc; cls = c; }
    }
    ec[(size_t)b * NPOS + (size_t)jj * NN + kk] = (unsigned char)cls;
  }

  // ---- node top-k over 128 g_node values ----
  for (int it = 0; it < kn; ++it) {
    float best = -3.0e38f; int bidx = 0x7FFFFFFF;
    bool absorbed = (nmj > 0.f) && (s_xcls[j] == DXC);
    if (absorbed) {
      bool taken = false;
      for (int q = 0; q < it; ++q) if (s_pickn[q] == j) { taken = true; break; }
      if (!taken) { best = g_node[(size_t)b * NN + j]; bidx = j; }
    }
    s_val[tid] = best; s_idx[tid] = bidx; __syncthreads();
    if (tid == 0) {
      float bb = s_val[0]; int bi = s_idx[0];
      for (int t = 1; t < NN; ++t)
        if (s_val[t] > bb || (s_val[t] == bb && s_idx[t] < bi)) { bb = s_val[t]; bi = s_idx[t]; }
      s_pickn[it] = bi;
    }
    __syncthreads();
  }

  // ---- node predictions ----
  for (int p = tid; p < kn; p += NN) {
    int jn = s_pickn[p];
    const float* lx = logits_x + ((size_t)b * NN + jn) * DXC;
    const float* gx = g_cat_x + ((size_t)b * NN + jn) * DXC;
    float m = lx[0];
    for (int c = 1; c < DXC; ++c) m = fmaxf(m, lx[c]);
    float den = 0.f;
    for (int c = 0; c < DXC; ++c) den += __expf(lx[c] - m);
    int cls = 0; float bsc = -3.0e38f;
    for (int c = 0; c < DXC; ++c) {
      float pc = __expf(lx[c] - m) / den;
      float sc = __logf(pc * __expf(s_dx[c]) + 1e-30f) + gx[c];
      if (sc > bsc) { bsc = sc; cls = c; }
    }
    xc[(size_t)b * NN + jn] = (unsigned char)cls;
  }
}

// ---------------------------------------------------------------------------
// K3: expand E classes to one-hot output (symmetric, diag zero, pair mask).
// ---------------------------------------------------------------------------
__global__ __launch_bounds__(128)
void ardg_k3_eout(const unsigned char* __restrict__ ec,
                  const unsigned char* __restrict__ nmask,
                  float* __restrict__ outE) {
  const int b = blockIdx.x, j = blockIdx.y, k = threadIdx.x;
  int jm = j < k ? j : k, km = j < k ? k : j;
  int cls = (j == k) ? -1 : (int)ec[(size_t)b * NPOS + (size_t)jm * NN + km];
  const unsigned char* nmb = nmask + (size_t)b * NN;
  if (!(nmb[j] && nmb[k])) cls = -1;
  float o0 = (cls == 0) ? 1.f : 0.f, o1 = (cls == 1) ? 1.f : 0.f;
  float o2 = (cls == 2) ? 1.f : 0.f, o3 = (cls == 3) ? 1.f : 0.f;
  float o4 = (cls == 4) ? 1.f : 0.f, o5 = (cls == 5) ? 1.f : 0.f;
  float* dst = outE + (((size_t)b * NN + j) * NN + k) * DE1;  // 24B stride, 8B aligned
  float2* d2 = reinterpret_cast<float2*>(dst);
  d2[0] = make_float2(o0, o1);
  d2[1] = make_float2(o2, o3);
  d2[2] = make_float2(o4, o5);
}

// ---------------------------------------------------------------------------
// K4: expand X classes to one-hot output (node_mask applied).
// ---------------------------------------------------------------------------
__global__ __launch_bounds__(128)
void ardg_k4_xout(const unsigned char* __restrict__ xc,
                  const unsigned char* __restrict__ nmask,
                  float* __restrict__ outX) {
  const int b = blockIdx.x, j = threadIdx.x;
  int cls = (int)xc[(size_t)b * NN + j];
  bool nm = nmask[(size_t)b * NN + j] != 0;
  float* dst = outX + ((size_t)b * NN + j) * DX1;
#pragma unroll
  for (int c = 0; c < DX1; ++c) dst[c] = (nm && c == cls) ? 1.f : 0.f;
}

// ---------------------------------------------------------------------------
extern "C" void kernel_launch(void* const* d_in, const int* in_sizes, int n_in,
                              void* d_out, int out_size, void* d_ws, size_t ws_size,
                              hipStream_t stream) {
  (void)in_sizes; (void)n_in; (void)out_size; (void)ws_size;
  const float* X        = (const float*)d_in[0];
  const float* E        = (const float*)d_in[1];
  const float* y        = (const float*)d_in[2];
  const unsigned char* nmask = (const unsigned char*)d_in[3]; // jnp bool = 1 byte
  const float* logits_x = (const float*)d_in[4];
  const float* logits_e = (const float*)d_in[5];
  const float* g_edge   = (const float*)d_in[6];
  const float* g_node   = (const float*)d_in[7];
  const float* g_cat_e  = (const float*)d_in[8];
  const float* g_cat_x  = (const float*)d_in[9];
  const float* wx       = (const float*)d_in[10];
  const float* we       = (const float*)d_in[11];
  const float* wy       = (const float*)d_in[12];
  const int*   step     = (const int*)d_in[13];

  // workspace layout (fully rewritten every call -> replay-safe)
  unsigned char* ec = (unsigned char*)d_ws;                    // B*16384 u8 = 4MB
  unsigned char* xc = ec + (size_t)BB * NPOS;                  // B*128 u8
  float* acc = (float*)(ec + (size_t)BB * NPOS + (size_t)BB * NN); // [B*SLICES][4] f32

  float* outX = (float*)d_out;
  float* outE = outX + NX_OUT;

  ardg_k1_epass<<<dim3(BB, SLICES), 256, 0, stream>>>(E, nmask, we, ec, acc);
  ardg_k2_finalize<<<BB, NN, 0, stream>>>(X, nmask, y, logits_x, logits_e,
                                          g_edge, g_node, g_cat_e, g_cat_x,
                                          wx, we, wy, step, acc, ec, xc);
  ardg_k3_eout<<<dim3(BB, NN), NN, 0, stream>>>(ec, nmask, outE);
  ardg_k4_xout<<<BB, NN, 0, stream>>>(xc, nmask, outX);
}